// ClassificationLoss_73804718014739
// MI455X (gfx1250) — compile-verified
//
#include <hip/hip_runtime.h>
#include <hip/hip_bf16.h>

// Problem constants (from reference setup_inputs)
#define B_IMG 16
#define N_PRED 25200
#define M_GT 64
#define C_CLS 80
#define BLOCK 128
#define NBLK ((N_PRED + BLOCK - 1) / BLOCK) // 197 blocks per image
#define NBLK_PAD 200                        // K padded to multiple of 4, zero-filled
#define TILE_DW (BLOCK * 85)                // 10880 dwords = 43.5 KB LDS tile
#define IOU_THR 0.4f

typedef __attribute__((ext_vector_type(2))) float v2f;
typedef __attribute__((ext_vector_type(8))) float v8f;
typedef __attribute__((ext_vector_type(4))) int v4i;

#if __has_builtin(__builtin_amdgcn_global_load_async_to_lds_b128)
#define USE_ASYNC_LDS 1
#else
#define USE_ASYNC_LDS 0
#endif

__device__ __forceinline__ float fast_exp2(float x) {
#if __has_builtin(__builtin_amdgcn_exp2f)
  return __builtin_amdgcn_exp2f(x);   // v_exp_f32
#else
  return exp2f(x);
#endif
}
__device__ __forceinline__ float fast_log2(float x) {
#if __has_builtin(__builtin_amdgcn_logf)
  return __builtin_amdgcn_logf(x);    // v_log_f32
#else
  return log2f(x);
#endif
}

#if USE_ASYNC_LDS
__device__ __forceinline__ void async_cp16(const float* g, float* l) {
  // Builtin signature (from hipcc diagnostic): (global int4*, local int4*, imm, imm)
  __builtin_amdgcn_global_load_async_to_lds_b128(
      (__attribute__((address_space(1))) v4i*)g,
      (__attribute__((address_space(3))) v4i*)l,
      /*offset=*/0, /*cpol=*/0);
}
__device__ __forceinline__ void wait_async0() {
#if __has_builtin(__builtin_amdgcn_s_wait_asynccnt)
  __builtin_amdgcn_s_wait_asynccnt(0);
#else
  asm volatile("s_wait_asynccnt 0" ::: "memory");
#endif
}
#endif

// ---------------------------------------------------------------------------
// Kernel 1: each block stages a tile of 128 predictions (85 f32 each) into LDS
// via GLOBAL_LOAD_ASYNC_TO_LDS_B128 (coalesced 16B/lane contiguous stream),
// then one thread per prediction computes: division-free IoU argmax against
// the 64 LDS-cached GT boxes, then exp2/log2-based log-softmax CE with the 80
// class scores held in VGPRs (LDS reads are bank-conflict-free: stride 85).
// Per-(image, block) partial sums go to d_ws.
// ---------------------------------------------------------------------------
__global__ __launch_bounds__(BLOCK) void loss_partial_kernel(
    const float* __restrict__ preds, const float* __restrict__ gts,
    float* __restrict__ ws_ce, float* __restrict__ ws_va) {
  __shared__ __align__(16) float tile[TILE_DW];
  __shared__ float s_gt[M_GT * 6];   // x1,y1,x2,y2,area,cls
  __shared__ float r_ce[BLOCK];
  __shared__ float r_va[BLOCK];

  const int tid = threadIdx.x;
  const int blk = blockIdx.x;
  const int b   = blockIdx.y;

  const int n0     = blk * BLOCK;
  const int npred  = (N_PRED - n0 < BLOCK) ? (N_PRED - n0) : BLOCK;
  const int nchunk = (npred * 85) >> 2;               // 16B chunks (always exact)
  const float* gbase = preds + ((size_t)b * N_PRED + n0) * 85;

  // Stage the prediction tile into LDS.
  for (int i = tid; i < nchunk; i += BLOCK) {
#if USE_ASYNC_LDS
    async_cp16(gbase + 4 * i, tile + 4 * i);
#else
    const float4 v = *reinterpret_cast<const float4*>(gbase + 4 * i);
    *reinterpret_cast<float4*>(tile + 4 * i) = v;
#endif
  }

  // GT boxes for this image (tiny; overlaps with the async stream).
  if (tid < M_GT) {
    const float* g = gts + ((size_t)b * M_GT + tid) * 5;
    const float x1 = g[0], y1 = g[1], x2 = g[2], y2 = g[3], cl = g[4];
    s_gt[tid * 6 + 0] = x1;
    s_gt[tid * 6 + 1] = y1;
    s_gt[tid * 6 + 2] = x2;
    s_gt[tid * 6 + 3] = y2;
    s_gt[tid * 6 + 4] = (x2 - x1) * (y2 - y1);
    s_gt[tid * 6 + 5] = cl;
  }

#if USE_ASYNC_LDS
  wait_async0();
#endif
  __syncthreads();

  float ce = 0.f, va = 0.f;
  if (tid < npred) {
    const float* pp = tile + tid * 85;   // LDS row, conflict-free banks
    const float px1 = pp[0], py1 = pp[1], px2 = pp[2], py2 = pp[3];
    const float ap = (px2 - px1) * (py2 - py1);

    // Division-free running argmax over IoU:
    //   iou_m > iou_best  <=>  inter_m * uni_best > inter_best * uni_m
    // (unions strictly positive). Init (bi=-1, bu=1) makes the first candidate
    // always win, matching jnp.argmax first-occurrence tie behavior.
    float bi = -1.f, bu = 1.f;
    int bidx = 0;
#pragma unroll 8
    for (int m = 0; m < M_GT; ++m) {
      const float gx1 = s_gt[m * 6 + 0];
      const float gy1 = s_gt[m * 6 + 1];
      const float gx2 = s_gt[m * 6 + 2];
      const float gy2 = s_gt[m * 6 + 3];
      const float ag  = s_gt[m * 6 + 4];
      float iw = fminf(px2, gx2) - fmaxf(px1, gx1);
      float ih = fminf(py2, gy2) - fmaxf(py1, gy1);
      iw = fmaxf(iw, 0.f);
      ih = fmaxf(ih, 0.f);
      const float inter = iw * ih;
      const float uni   = ap + ag - inter;
      const bool better = inter * bu > bi * uni;
      bi   = better ? inter : bi;
      bu   = better ? uni   : bu;
      bidx = better ? m     : bidx;
    }
    va = (bi >= IOU_THR * bu) ? 1.f : 0.f;
    const int label = (int)s_gt[bidx * 6 + 5];

    // log-softmax CE over the 80 class scores (skip obj at [4]).
    const float* sp = pp + 5;
    float sc[C_CLS];
#pragma unroll
    for (int c = 0; c < C_CLS; ++c) sc[c] = sp[c];
    float mx = sc[0];
#pragma unroll
    for (int c = 1; c < C_CLS; ++c) mx = fmaxf(mx, sc[c]);
    const float L2E = 1.4426950408889634f;
    float S = 0.f;
#pragma unroll
    for (int c = 0; c < C_CLS; ++c) S += fast_exp2((sc[c] - mx) * L2E);
    const float s_lab = sp[label];  // dynamic index -> cheap ds_load
    const float lse = mx + fast_log2(S) * 0.6931471805599453f;
    ce = (lse - s_lab) * va;
  }

  // Deterministic block tree reduction.
  r_ce[tid] = ce;
  r_va[tid] = va;
  __syncthreads();
  for (int s = BLOCK / 2; s > 0; s >>= 1) {
    if (tid < s) {
      r_ce[tid] += r_ce[tid + s];
      r_va[tid] += r_va[tid + s];
    }
    __syncthreads();
  }
  if (tid == 0) {
    ws_ce[b * NBLK_PAD + blk] = r_ce[0];
    ws_va[b * NBLK_PAD + blk] = r_va[0];
  }
}

// ---------------------------------------------------------------------------
// Kernel 2: single wave32. Row-sums of the (zero-padded) 16 x 200 partial
// matrices via V_WMMA_F32_16X16X4_F32 with an all-ones B (D = A*1 + C gives
// exact, deterministic f32 row sums on the matrix pipe), then the final
// masked mean over images.
// ---------------------------------------------------------------------------
__global__ __launch_bounds__(32) void finalize_kernel(
    const float* __restrict__ ws_ce, const float* __restrict__ ws_va,
    float* __restrict__ out) {
  const int lane = threadIdx.x;
  __shared__ float s_ce[B_IMG];
  __shared__ float s_va[B_IMG];

#if __has_builtin(__builtin_amdgcn_wmma_f32_16x16x4_f32)
  // A layout (16x4 f32): lanes 0-15 hold {K=0,K=1} for M=lane;
  //                      lanes 16-31 hold {K=2,K=3} for M=lane-16.
  const int row = lane & 15;
  const int kh  = (lane >> 4) << 1;  // 0 or 2
  v8f acc_ce = {};
  v8f acc_va = {};
  v2f ones;
  ones.x = 1.f;
  ones.y = 1.f;  // B == all-ones regardless of B's lane layout
  const float2* pc = reinterpret_cast<const float2*>(ws_ce);
  const float2* pv = reinterpret_cast<const float2*>(ws_va);
  for (int k0 = 0; k0 < NBLK_PAD; k0 += 4) {
    const int idx2 = row * (NBLK_PAD / 2) + ((k0 + kh) >> 1);  // 8B-aligned pair
    const float2 tc = pc[idx2];
    const float2 tv = pv[idx2];
    v2f a_ce, a_va;
    a_ce.x = tc.x; a_ce.y = tc.y;
    a_va.x = tv.x; a_va.y = tv.y;
    acc_ce = __builtin_amdgcn_wmma_f32_16x16x4_f32(
        false, a_ce, false, ones, (short)0, acc_ce, false, false);
    acc_va = __builtin_amdgcn_wmma_f32_16x16x4_f32(
        false, a_va, false, ones, (short)0, acc_va, false, false);
  }
  // D layout: VGPR r, lane 0  -> (M=r,   N=0); lane 16 -> (M=r+8, N=0).
  if (lane == 0) {
#pragma unroll
    for (int r = 0; r < 8; ++r) { s_ce[r] = acc_ce[r]; s_va[r] = acc_va[r]; }
  }
  if (lane == 16) {
#pragma unroll
    for (int r = 0; r < 8; ++r) { s_ce[8 + r] = acc_ce[r]; s_va[8 + r] = acc_va[r]; }
  }
#else
  if (lane < B_IMG) {
    float sc_ = 0.f, sv_ = 0.f;
    for (int k = 0; k < NBLK; ++k) {
      sc_ += ws_ce[lane * NBLK_PAD + k];
      sv_ += ws_va[lane * NBLK_PAD + k];
    }
    s_ce[lane] = sc_;
    s_va[lane] = sv_;
  }
#endif
  __syncthreads();
  if (lane == 0) {
    float acc = 0.f;
    for (int b = 0; b < B_IMG; ++b) acc += s_ce[b] / fmaxf(s_va[b], 1.f);
    out[0] = acc * (1.0f / B_IMG);
  }
}

extern "C" void kernel_launch(void* const* d_in, const int* in_sizes, int n_in,
                              void* d_out, int out_size, void* d_ws, size_t ws_size,
                              hipStream_t stream) {
  (void)in_sizes; (void)n_in; (void)out_size; (void)ws_size;
  const float* preds = (const float*)d_in[0];
  const float* gts   = (const float*)d_in[1];
  float* out = (float*)d_out;

  float* ws_ce = (float*)d_ws;                     // 16 x 200 f32 (zero-padded)
  float* ws_va = ws_ce + (size_t)B_IMG * NBLK_PAD; // 16 x 200 f32 (zero-padded)

  // Zero the padded partial matrices so kernel 2 needs no bounds checks.
  (void)hipMemsetAsync(d_ws, 0, (size_t)2 * B_IMG * NBLK_PAD * sizeof(float), stream);

  dim3 grid(NBLK, B_IMG);
  loss_partial_kernel<<<grid, BLOCK, 0, stream>>>(preds, gts, ws_ce, ws_va);
  finalize_kernel<<<1, 32, 0, stream>>>(ws_ce, ws_va, out);
}